// HeatConvLayer_34437047779554
// MI455X (gfx1250) — compile-verified
//
#include <hip/hip_runtime.h>

typedef __attribute__((ext_vector_type(2))) float v2f;
typedef __attribute__((ext_vector_type(4))) float v4f;
typedef __attribute__((ext_vector_type(8))) float v8f;

#define N_NODES 100000
#define N_EDGES 640000
#define DIM     128
#define NLAYERS 4
#define AS      (DIM + 4)   // padded LDS row stride (132): conflict-free strided column reads

// ---------------------------------------------------------------------------
// One-time zero of the aggregation buffer (layer 0 only; later layers are
// re-zeroed inside mlp_fused_kernel by the block that owns the rows).
// ---------------------------------------------------------------------------
__global__ __launch_bounds__(256) void zero_h0_kernel(float* __restrict__ h0) {
  int idx = (blockIdx.x * 256 + threadIdx.x) * 4;
  if (idx >= N_NODES * DIM) return;
  v4f z = {};
  *(v4f*)(h0 + idx) = z;
}

// ---------------------------------------------------------------------------
// For each edge e: h0[dst[e]] += relu(x[src[e]] + edge_attr[e])
// 32 threads per edge, 4 features per thread (float4).
// edge_attr is streamed (327MB, read once per layer) -> non-temporal so the
// L2-resident working set (x, h0, ping-pong buffers: ~150MB of 192MB L2)
// is not thrashed.
// ---------------------------------------------------------------------------
__global__ __launch_bounds__(256) void edge_msg_kernel(
    const float* __restrict__ xin, const int* __restrict__ ei,
    const float* __restrict__ ea, float* __restrict__ h0) {
  unsigned tid = blockIdx.x * 256 + threadIdx.x;
  unsigned e = tid >> 5;
  if (e >= N_EDGES) return;
  unsigned c4 = (tid & 31u) << 2;
  int s = ei[e];
  int d = ei[N_EDGES + e];
  v4f av = __builtin_nontemporal_load((const v4f*)(ea + (size_t)e * DIM + c4));
  v4f xv = *(const v4f*)(xin + (size_t)s * DIM + c4);
  float m0 = fmaxf(xv.x + av.x, 0.0f);
  float m1 = fmaxf(xv.y + av.y, 0.0f);
  float m2 = fmaxf(xv.z + av.z, 0.0f);
  float m3 = fmaxf(xv.w + av.w, 0.0f);
  float* hp = h0 + (size_t)d * DIM + c4;
  atomicAdd(hp + 0, m0);
  atomicAdd(hp + 1, m1);
  atomicAdd(hp + 2, m2);
  atomicAdd(hp + 3, m3);
}

// ---------------------------------------------------------------------------
// Fused: a = (1+eps)*x + agg; t = relu(a @ W1 + b1); h = t @ W2 + b2;
//        v = BN(mask*h + x);  out = is_final ? xorig + relu(v) : v
// Also re-zeroes this block's h0 rows for the next layer (do_zero).
// 128 rows per block; 8 waves, each wave owns a 16-row WMMA strip.
// W (64KB) + padded A tile (66KB) staged in LDS (CDNA5: 320KB/WGP).
// GEMM core: v_wmma_f32_16x16x4_f32, K-loop of 32, 8 N-tiles per wave.
// ---------------------------------------------------------------------------
__global__ __launch_bounds__(256) void mlp_fused_kernel(
    float* __restrict__ h0, const float* __restrict__ xin,
    const float* __restrict__ eps, int layer,
    const float* __restrict__ W1, const float* __restrict__ b1,
    const float* __restrict__ W2, const float* __restrict__ b2,
    const float* __restrict__ mask,
    const float* __restrict__ gamma, const float* __restrict__ beta,
    const float* __restrict__ rmean, const float* __restrict__ rvar,
    const float* __restrict__ xorig, int is_final, int do_zero,
    float* __restrict__ xout) {
  __shared__ float sW[DIM * DIM];   // 64 KB weight stage
  __shared__ float sA[DIM * AS];    // 66 KB activation tile (padded rows)

  const int t = threadIdx.x;
  const int m0 = blockIdx.x * DIM;
  const float epss = 1.0f + eps[layer];   // uniform scalar load

  // Stage A tile: (1+eps)*x + agg, zero-padded past N. Stage W1.
  for (int idx = t * 4; idx < DIM * DIM; idx += 256 * 4) {
    int row = idx >> 7, col = idx & (DIM - 1);
    int n = m0 + row;
    v4f a = {};
    if (n < N_NODES) {
      v4f agg = *(const v4f*)(h0 + (size_t)n * DIM + col);
      v4f xv  = *(const v4f*)(xin + (size_t)n * DIM + col);
      a.x = epss * xv.x + agg.x;
      a.y = epss * xv.y + agg.y;
      a.z = epss * xv.z + agg.z;
      a.w = epss * xv.w + agg.w;
    }
    *(v4f*)(sA + row * AS + col) = a;
    *(v4f*)(sW + idx) = *(const v4f*)(W1 + idx);
  }
  __syncthreads();

  // This block owns rows [m0, m0+128): zero its h0 slice for the next layer's
  // edge atomics (next edge kernel is a later launch on the same stream).
  // Stores overlap with GEMM1 compute below.
  if (do_zero) {
    v4f z = {};
    for (int idx = t * 4; idx < DIM * DIM; idx += 256 * 4) {
      int n = m0 + (idx >> 7);
      if (n < N_NODES) *(v4f*)(h0 + (size_t)n * DIM + (idx & (DIM - 1))) = z;
    }
  }

  const int lane = t & 31;
  const int wave = t >> 5;
  const int lrow = lane & 15;
  const int half = lane >> 4;
  const int r0 = wave * 16;          // this wave's row strip within the tile

  v8f acc[8];
#pragma unroll
  for (int nt = 0; nt < 8; nt++) { v8f z = {}; acc[nt] = z; }

  // ---- GEMM1: acc = A @ W1 -----------------------------------------------
  for (int k = 0; k < 32; k++) {
    const int kk = 4 * k + 2 * half;                 // A: lanes 0-15 K=0,1; 16-31 K=2,3
    v2f a = *(const v2f*)(sA + (r0 + lrow) * AS + kk);
#pragma unroll
    for (int nt = 0; nt < 8; nt++) {
      v2f b;
      b.x = sW[(kk + 0) * DIM + nt * 16 + lrow];     // B: N striped across lanes
      b.y = sW[(kk + 1) * DIM + nt * 16 + lrow];
      acc[nt] = __builtin_amdgcn_wmma_f32_16x16x4_f32(
          false, a, false, b, (short)0, acc[nt], false, false);
    }
  }
  __syncthreads();  // all waves done reading sA / sW

  // relu(acc + b1) -> sA (D layout: vgpr r -> row r + 8*half, col nt*16+lrow)
#pragma unroll
  for (int nt = 0; nt < 8; nt++) {
    float bb = b1[nt * 16 + lrow];
#pragma unroll
    for (int r = 0; r < 8; r++) {
      float v = fmaxf(acc[nt][r] + bb, 0.0f);
      sA[(r0 + r + 8 * half) * AS + nt * 16 + lrow] = v;
    }
    v8f z = {};
    acc[nt] = z;
  }
  // Restage W2.
  for (int idx = t * 4; idx < DIM * DIM; idx += 256 * 4)
    *(v4f*)(sW + idx) = *(const v4f*)(W2 + idx);
  __syncthreads();

  // ---- GEMM2: acc = relu(...) @ W2 ---------------------------------------
  for (int k = 0; k < 32; k++) {
    const int kk = 4 * k + 2 * half;
    v2f a = *(const v2f*)(sA + (r0 + lrow) * AS + kk);
#pragma unroll
    for (int nt = 0; nt < 8; nt++) {
      v2f b;
      b.x = sW[(kk + 0) * DIM + nt * 16 + lrow];
      b.y = sW[(kk + 1) * DIM + nt * 16 + lrow];
      acc[nt] = __builtin_amdgcn_wmma_f32_16x16x4_f32(
          false, a, false, b, (short)0, acc[nt], false, false);
    }
  }

  // ---- Epilogue: bias, mask-residual, BN, optional final residual --------
#pragma unroll
  for (int nt = 0; nt < 8; nt++) {
    const int col = nt * 16 + lrow;
    const float bb = b2[col];
    const float sc = gamma[col] * rsqrtf(rvar[col] + 1e-5f);
    const float sh = beta[col] - rmean[col] * sc;
#pragma unroll
    for (int r = 0; r < 8; r++) {
      const int n = m0 + r0 + r + 8 * half;
      if (n < N_NODES) {
        float h = acc[nt][r] + bb;
        float v = mask[n] * h + xin[(size_t)n * DIM + col];
        v = v * sc + sh;
        if (is_final) {
          float xo = xorig[(size_t)n * DIM + col];
          xout[(size_t)n * DIM + col] = xo + fmaxf(v, 0.0f);
        } else {
          xout[(size_t)n * DIM + col] = v;
        }
      }
    }
  }
}

// ---------------------------------------------------------------------------
extern "C" void kernel_launch(void* const* d_in, const int* in_sizes, int n_in,
                              void* d_out, int out_size, void* d_ws, size_t ws_size,
                              hipStream_t stream) {
  const float* x     = (const float*)d_in[0];
  const int*   ei    = (const int*)d_in[1];
  const float* ea    = (const float*)d_in[2];
  const float* masks = (const float*)d_in[3];
  const float* W1    = (const float*)d_in[4];
  const float* b1    = (const float*)d_in[5];
  const float* W2    = (const float*)d_in[6];
  const float* b2    = (const float*)d_in[7];
  const float* eps   = (const float*)d_in[8];
  const float* gamma = (const float*)d_in[9];
  const float* beta  = (const float*)d_in[10];
  const float* rmean = (const float*)d_in[11];
  const float* rvar  = (const float*)d_in[12];
  float* out = (float*)d_out;
  float* ws  = (float*)d_ws;

  float* h0 = ws;                                    // N*D floats (agg buffer)
  float* xbuf[2];
  xbuf[0] = ws + (size_t)N_NODES * DIM;              // N*D floats
  xbuf[1] = out;                                     // d_out doubles as ping-pong buf

  const int elem_blocks = (N_NODES * DIM / 4 + 255) / 256;           // 12500
  const int edge_blocks = (int)(((size_t)N_EDGES * 32 + 255) / 256); // 80000
  const int mlp_blocks  = (N_NODES + DIM - 1) / DIM;                 // 782

  zero_h0_kernel<<<elem_blocks, 256, 0, stream>>>(h0);

  const float* xcur = x;
  for (int c = 0; c < NLAYERS; c++) {
    float* xnext = xbuf[c & 1];                      // c=0:b0, 1:out, 2:b0, 3:out
    edge_msg_kernel<<<edge_blocks, 256, 0, stream>>>(xcur, ei, ea, h0);
    mlp_fused_kernel<<<mlp_blocks, 256, 0, stream>>>(
        h0, xcur, eps, c,
        W1 + (size_t)c * DIM * DIM, b1 + c * DIM,
        W2 + (size_t)c * DIM * DIM, b2 + c * DIM,
        masks + (size_t)c * N_NODES,
        gamma + c * DIM, beta + c * DIM, rmean + c * DIM, rvar + c * DIM,
        x, /*is_final=*/(c == NLAYERS - 1), /*do_zero=*/(c < NLAYERS - 1),
        xnext);
    xcur = xnext;                                    // after c=3, xcur == out
  }
}